// ConcatRead_35588099014909
// MI455X (gfx1250) — compile-verified
//
#include <hip/hip_runtime.h>
#include <hip/hip_bf16.h>

typedef __attribute__((ext_vector_type(2))) float v2f;
typedef __attribute__((ext_vector_type(8))) float v8f;

#define HH     256
#define HH2    512
#define HALFD  128
#define MPAD   132   // padded row stride: 132 mod 64 = 4 -> rows start 4 banks apart
#define BSZ    64
#define LSEQ   512
#define VOCAB  32000

// D = A(16x4,f32) * B(4x16,f32) + C(16x16,f32), fp32-exact WMMA.
__device__ __forceinline__ v8f wmma_f32(v2f a, v2f b, v8f c) {
  return __builtin_amdgcn_wmma_f32_16x16x4_f32(
      /*neg_a=*/false, a, /*neg_b=*/false, b,
      /*c_mod=*/(short)0, c, /*reuse_a=*/false, /*reuse_b=*/false);
}

// ---------------------------------------------------------------------------
// Kernel 1: embed-gather + FF + residual + LayerNorm + fused k-projections.
// One block = 16 token rows, 128 threads = 4 waves. All GEMMs via WMMA f32.
// Writes ks, ke: (B*L, 128) row-major; scan uses rows t=0..510, query row 511.
// ---------------------------------------------------------------------------
__global__ __launch_bounds__(128) void encode_kernel(
    const int*   __restrict__ seq,   const float* __restrict__ embed,
    const float* __restrict__ W1,    const float* __restrict__ b1,
    const float* __restrict__ W2,    const float* __restrict__ b2,
    const float* __restrict__ ln_g,  const float* __restrict__ ln_b,
    const float* __restrict__ Wsem,  const float* __restrict__ bsem,
    const float* __restrict__ Wepi,  const float* __restrict__ bepi,
    float* __restrict__ ks, float* __restrict__ ke)
{
  __shared__ float sE[16 * HH];    // 16KB: embeddings, later LN'd rows
  __shared__ float sF1[16 * HH2];  // 32KB: relu(E@W1+b1)
  __shared__ float sX[16 * HH];    // 16KB: residual x = e + ff

  const int tid  = threadIdx.x;
  const int lane = tid & 31;
  const int wave = tid >> 5;
  const int row0 = blockIdx.x * 16;

  // lane -> (n column within 16-wide tile, K sub-offset) per ISA 16x16x4 layout
  const int lane_n = (lane < 16) ? lane : (lane - 16);
  const int koff   = (lane < 16) ? 0 : 2;

  // ---- gather 16 embedding rows into LDS (float4 vectorized) ----
  {
    float4* sE4 = reinterpret_cast<float4*>(sE);
    for (int idx = tid; idx < 16 * (HH / 4); idx += 128) {
      int r = idx >> 6, c = idx & 63;
      int tok = seq[row0 + r];
      const float4* er = reinterpret_cast<const float4*>(embed + (size_t)tok * HH);
      sE4[r * (HH / 4) + c] = er[c];
    }
  }
  __syncthreads();

  // ---- ff1 = relu(E @ W1 + b1): 16x512, 32 n-tiles split over 4 waves ----
  for (int nt = wave; nt < 32; nt += 4) {
    const int n0 = nt * 16;
    float bv = b1[n0 + lane_n];
    v8f acc;
    #pragma unroll
    for (int r = 0; r < 8; ++r) acc[r] = bv;
    for (int k0 = 0; k0 < HH; k0 += 4) {
      v2f a, b;
      a.x = sE[lane_n * HH + k0 + koff];
      a.y = sE[lane_n * HH + k0 + koff + 1];
      b.x = W1[(size_t)(k0 + koff)     * HH2 + n0 + lane_n];
      b.y = W1[(size_t)(k0 + koff + 1) * HH2 + n0 + lane_n];
      acc = wmma_f32(a, b, acc);
    }
    #pragma unroll
    for (int r = 0; r < 8; ++r) {
      int m = (lane < 16) ? r : (r + 8);
      float v = acc[r];
      sF1[m * HH2 + n0 + lane_n] = v > 0.f ? v : 0.f;
    }
  }
  __syncthreads();

  // ---- x = E + (F1 @ W2 + b2): 16x256, 16 n-tiles ----
  for (int nt = wave; nt < 16; nt += 4) {
    const int n0 = nt * 16;
    float bv = b2[n0 + lane_n];
    v8f acc;
    #pragma unroll
    for (int r = 0; r < 8; ++r) acc[r] = bv;
    for (int k0 = 0; k0 < HH2; k0 += 4) {
      v2f a, b;
      a.x = sF1[lane_n * HH2 + k0 + koff];
      a.y = sF1[lane_n * HH2 + k0 + koff + 1];
      b.x = W2[(size_t)(k0 + koff)     * HH + n0 + lane_n];
      b.y = W2[(size_t)(k0 + koff + 1) * HH + n0 + lane_n];
      acc = wmma_f32(a, b, acc);
    }
    #pragma unroll
    for (int r = 0; r < 8; ++r) {
      int m = (lane < 16) ? r : (r + 8);
      int c = n0 + lane_n;
      sX[m * HH + c] = sE[m * HH + c] + acc[r];
    }
  }
  __syncthreads();

  // ---- LayerNorm: 8 lanes per row, shuffle reductions; result into sE ----
  {
    const int r = tid >> 3;        // row 0..15
    const int s = tid & 7;         // sub-lane 0..7 (aligned 8-groups in a wave)
    float lsum = 0.f;
    for (int c = s; c < HH; c += 8) lsum += sX[r * HH + c];
    lsum += __shfl_xor(lsum, 1);
    lsum += __shfl_xor(lsum, 2);
    lsum += __shfl_xor(lsum, 4);
    const float mu = lsum * (1.f / (float)HH);
    float lvar = 0.f;
    for (int c = s; c < HH; c += 8) { float d = sX[r * HH + c] - mu; lvar += d * d; }
    lvar += __shfl_xor(lvar, 1);
    lvar += __shfl_xor(lvar, 2);
    lvar += __shfl_xor(lvar, 4);
    const float inv = rsqrtf(lvar * (1.f / (float)HH) + 1e-5f);
    for (int c = s; c < HH; c += 8)
      sE[r * HH + c] = (sX[r * HH + c] - mu) * inv * ln_g[c] + ln_b[c];
  }
  __syncthreads();

  // ---- fused projections: ks = h@Wsem+bsem, ke = h@Wepi+bepi (16x128 each) ----
  for (int nt = wave; nt < 16; nt += 4) {
    const int matsel = nt >> 3;            // 0 -> sem, 1 -> epi
    const int n0 = (nt & 7) * 16;
    const float* W    = matsel ? Wepi : Wsem;
    const float* bias = matsel ? bepi : bsem;
    float*       dst  = matsel ? ke   : ks;
    float bv = bias[n0 + lane_n];
    v8f acc;
    #pragma unroll
    for (int r = 0; r < 8; ++r) acc[r] = bv;
    for (int k0 = 0; k0 < HH; k0 += 4) {
      v2f a, b;
      a.x = sE[lane_n * HH + k0 + koff];
      a.y = sE[lane_n * HH + k0 + koff + 1];
      b.x = W[(size_t)(k0 + koff)     * HALFD + n0 + lane_n];
      b.y = W[(size_t)(k0 + koff + 1) * HALFD + n0 + lane_n];
      acc = wmma_f32(a, b, acc);
    }
    #pragma unroll
    for (int r = 0; r < 8; ++r) {
      int m = (lane < 16) ? r : (r + 8);
      dst[(size_t)(row0 + m) * HALFD + n0 + lane_n] = acc[r];
    }
  }
}

// ---------------------------------------------------------------------------
// Kernel 2: sequential delta-rule scan. One workgroup per (batch, matrix);
// M (128x132-padded f32 ~= 67.6KB) resident in LDS for all 511 steps.
// 512 threads: each owns a quarter-row of M (exclusive read+write), so the
// matvec->update chain needs NO LDS traffic for vp/dv: the 4 lanes owning a
// row combine partials with shuffles and recompute dv redundantly.
// kv/red are double-buffered on t&1 -> ONE barrier per step: by the time all
// threads pass step t+1's publish barrier, every step-t read has completed,
// so step t+2 may safely overwrite the same buffer.
// Row stride 132 (= 4 mod 64 banks) lets quarter-rows be accessed as float4
// (ds_load_b128/ds_store_b128) with rows landing on disjoint bank groups.
// ---------------------------------------------------------------------------
__global__ __launch_bounds__(512) void scan_kernel(
    const float* __restrict__ ks, const float* __restrict__ ke,
    float* __restrict__ cbuf)
{
  __shared__ float M[HALFD * MPAD];    // ~67.6KB state (padded rows)
  __shared__ float kv[2][HALFD];       // ping-pong k vector
  __shared__ float red[2][4];          // ping-pong ||k||^2 wave partials

  const int tid = threadIdx.x;
  const int b   = blockIdx.x;          // batch
  const int typ = blockIdx.y;          // 0 = M_s, 1 = M_e
  const float* kbuf = typ ? ke : ks;

  const int i  = tid >> 2;             // row 0..127
  const int q  = tid & 3;              // quarter 0..3
  const int jb = q * 32;
  float4* Mq = reinterpret_cast<float4*>(&M[i * MPAD + jb]); // 16B aligned

  for (int idx = tid; idx < HALFD * MPAD; idx += 512) M[idx] = 0.f;
  __syncthreads();

  const float* kr = kbuf + (size_t)b * LSEQ * HALFD;
  for (int t = 0; t < LSEQ - 1; ++t) {
    const int buf = t & 1;
    if (tid < HALFD) {
      float v = kr[(size_t)t * HALFD + tid];
      kv[buf][tid] = v;
      float p = v * v;                 // wave-level ||k||^2 partials
      p += __shfl_xor(p, 16);
      p += __shfl_xor(p, 8);
      p += __shfl_xor(p, 4);
      p += __shfl_xor(p, 2);
      p += __shfl_xor(p, 1);
      if ((tid & 31) == 0) red[buf][tid >> 5] = p;
      // pull next step's k row toward the WGP while we compute this one
      __builtin_prefetch(&kr[(size_t)(t + 1) * HALFD + tid], 0, 0);
    }
    __syncthreads();                   // publish kv/red for this step
    const float denom = red[buf][0] + red[buf][1] + red[buf][2] + red[buf][3] + 1e-6f;
    const float4* kq = reinterpret_cast<const float4*>(&kv[buf][jb]);

    // vp_i = (M @ k)[i] : 8 float4 MACs on own quarter-row, combine 4 lanes
    float sum = 0.f;
    #pragma unroll
    for (int f = 0; f < 8; ++f) {
      float4 m = Mq[f], k4 = kq[f];
      sum += m.x * k4.x + m.y * k4.y + m.z * k4.z + m.w * k4.w;
    }
    sum += __shfl_xor(sum, 1);
    sum += __shfl_xor(sum, 2);         // full vp in all 4 quarter-lanes

    const float sc  = typ ? (float)(t + 1) * (1.0f / (float)LSEQ) : 1.0f;
    const float dvi = (kv[buf][i] - sum / denom) * sc;

    #pragma unroll
    for (int f = 0; f < 8; ++f) {
      float4 m = Mq[f], k4 = kq[f];
      m.x += dvi * k4.x; m.y += dvi * k4.y;
      m.z += dvi * k4.z; m.w += dvi * k4.w;
      Mq[f] = m;
    }
    // no trailing barrier: next step's publish barrier is the fence
  }

  // readout with the query projection (row L-1 of kbuf)
  {
    const int buf = (LSEQ - 1) & 1;
    if (tid < HALFD) kv[buf][tid] = kr[(size_t)(LSEQ - 1) * HALFD + tid];
    __syncthreads();
    const float4* kq = reinterpret_cast<const float4*>(&kv[buf][jb]);
    float sum = 0.f;
    #pragma unroll
    for (int f = 0; f < 8; ++f) {
      float4 m = Mq[f], k4 = kq[f];
      sum += m.x * k4.x + m.y * k4.y + m.z * k4.z + m.w * k4.w;
    }
    sum += __shfl_xor(sum, 1);
    sum += __shfl_xor(sum, 2);
    if (q == 0)
      cbuf[(size_t)b * (2 * HALFD) + typ * HALFD + i] = sum;
  }
}

// ---------------------------------------------------------------------------
// Kernel 3: out = c (64x256) @ Wout (256x32000) + bout. One wave per n-tile
// computes ALL four 16-row m-tiles, so each Wout fragment is loaded once and
// reused 4x -> Wout streamed from HBM exactly once (~33MB @ 23.3TB/s).
// ---------------------------------------------------------------------------
__global__ __launch_bounds__(32) void out_kernel(
    const float* __restrict__ cbuf, const float* __restrict__ Wout,
    const float* __restrict__ bout, float* __restrict__ out)
{
  const int lane = threadIdx.x;
  const int n0 = blockIdx.x * 16;      // 2000 tiles over V
  const int lane_n = (lane < 16) ? lane : (lane - 16);
  const int koff   = (lane < 16) ? 0 : 2;

  float bv = bout[n0 + lane_n];
  v8f acc0, acc1, acc2, acc3;
  #pragma unroll
  for (int r = 0; r < 8; ++r) { acc0[r] = bv; acc1[r] = bv; acc2[r] = bv; acc3[r] = bv; }

  for (int k0 = 0; k0 < 2 * HALFD; k0 += 4) {
    v2f bfr;
    bfr.x = Wout[(size_t)(k0 + koff)     * VOCAB + n0 + lane_n];
    bfr.y = Wout[(size_t)(k0 + koff + 1) * VOCAB + n0 + lane_n];
    v2f a0, a1, a2, a3;
    const size_t cb = (size_t)lane_n * (2 * HALFD) + k0 + koff;
    a0.x = cbuf[cb];                       a0.y = cbuf[cb + 1];
    a1.x = cbuf[cb + 16 * 2 * HALFD];      a1.y = cbuf[cb + 16 * 2 * HALFD + 1];
    a2.x = cbuf[cb + 32 * 2 * HALFD];      a2.y = cbuf[cb + 32 * 2 * HALFD + 1];
    a3.x = cbuf[cb + 48 * 2 * HALFD];      a3.y = cbuf[cb + 48 * 2 * HALFD + 1];
    acc0 = wmma_f32(a0, bfr, acc0);
    acc1 = wmma_f32(a1, bfr, acc1);
    acc2 = wmma_f32(a2, bfr, acc2);
    acc3 = wmma_f32(a3, bfr, acc3);
  }
  #pragma unroll
  for (int r = 0; r < 8; ++r) {
    int m = (lane < 16) ? r : (r + 8);
    out[(size_t)(m)      * VOCAB + n0 + lane_n] = acc0[r];
    out[(size_t)(m + 16) * VOCAB + n0 + lane_n] = acc1[r];
    out[(size_t)(m + 32) * VOCAB + n0 + lane_n] = acc2[r];
    out[(size_t)(m + 48) * VOCAB + n0 + lane_n] = acc3[r];
  }
}

extern "C" void kernel_launch(void* const* d_in, const int* in_sizes, int n_in,
                              void* d_out, int out_size, void* d_ws, size_t ws_size,
                              hipStream_t stream) {
  (void)in_sizes; (void)n_in; (void)out_size; (void)ws_size;
  const int*   seq   = (const int*)  d_in[0];
  const float* embed = (const float*)d_in[1];
  const float* W1    = (const float*)d_in[2];
  const float* b1    = (const float*)d_in[3];
  const float* W2    = (const float*)d_in[4];
  const float* b2    = (const float*)d_in[5];
  const float* ln_g  = (const float*)d_in[6];
  const float* ln_b  = (const float*)d_in[7];
  const float* Wsem  = (const float*)d_in[8];
  const float* bsem  = (const float*)d_in[9];
  const float* Wepi  = (const float*)d_in[10];
  const float* bepi  = (const float*)d_in[11];
  const float* Wout  = (const float*)d_in[12];
  const float* bout  = (const float*)d_in[13];
  float* out = (float*)d_out;

  // workspace: ks (B*L x 128) | ke (B*L x 128) | c (B x 256)  ~= 33.6 MB
  float* ks = (float*)d_ws;
  float* ke = ks + (size_t)BSZ * LSEQ * HALFD;
  float* cb = ke + (size_t)BSZ * LSEQ * HALFD;

  encode_kernel<<<(BSZ * LSEQ) / 16, 128, 0, stream>>>(
      seq, embed, W1, b1, W2, b2, ln_g, ln_b, Wsem, bsem, Wepi, bepi, ks, ke);
  scan_kernel<<<dim3(BSZ, 2), 512, 0, stream>>>(ks, ke, cb);
  out_kernel<<<VOCAB / 16, 32, 0, stream>>>(cb, Wout, bout, out);
}